// SparseAvgPool_48584670053118
// MI455X (gfx1250) — compile-verified
//
#include <hip/hip_runtime.h>

#define WAVES 8
#define NPAIR 4
#define CCH   64   // channels, fixed by the reference (C = 64)

#if defined(__HIP_DEVICE_COMPILE__) && \
    __has_builtin(__builtin_amdgcn_global_load_async_to_lds_b64) && \
    __has_builtin(__builtin_amdgcn_s_wait_asynccnt)
#define USE_ASYNC 1
#else
#define USE_ASYNC 0
#endif

#if USE_ASYNC
typedef int v2i __attribute__((ext_vector_type(2)));
typedef __attribute__((address_space(1))) v2i* gv2i_p;   // global (prints as __device__)
typedef __attribute__((address_space(3))) v2i* lv2i_p;   // LDS    (prints as __shared__)
#endif

// ---------------- zero-init sums + counts ----------------
__global__ void spavg_zero_kernel(float4* __restrict__ out4, int n4,
                                  float* __restrict__ cnt, int nc) {
    const int stride = gridDim.x * blockDim.x;
    for (int i = blockIdx.x * blockDim.x + threadIdx.x; i < n4; i += stride)
        out4[i] = make_float4(0.f, 0.f, 0.f, 0.f);
    for (int i = blockIdx.x * blockDim.x + threadIdx.x; i < nc; i += stride)
        cnt[i] = 0.f;
}

// ---------------- scatter-sum ----------------
__device__ __forceinline__ void spavg_process_pair(const float* __restrict__ src2,
                                                   bool valid, int lane, int oidx,
                                                   float* __restrict__ out,
                                                   float* __restrict__ counts) {
    if (valid) {
        const float2 v = *(const float2*)src2;            // ds_load_b64 (from LDS stage)
        float* dst = out + (size_t)oidx * CCH + lane * 2;
        atomicAdd(dst,     v.x);                          // global_atomic_add_f32 (no-return)
        atomicAdd(dst + 1, v.y);
        if (lane == 0) atomicAdd(counts + oidx, 1.0f);
    }
}

__global__ void spavg_scatter_kernel(const float* __restrict__ feat,
                                     const int*   __restrict__ pairs,
                                     const int*   __restrict__ nums,
                                     float*       __restrict__ out,
                                     float*       __restrict__ counts,
                                     int P) {
    __shared__ float smem[WAVES * NPAIR * CCH];          // 8 KB / block
    const int lane = threadIdx.x & 31;                   // wave32
    const int wave = threadIdx.x >> 5;
    const int k    = blockIdx.y;
    const int n    = nums[k];                            // num valid pairs for offset k (< P)
    const int p0   = (blockIdx.x * WAVES + wave) * NPAIR;
    const int* __restrict__ pk = pairs + (size_t)k * 2 * P;
    float* wsm = smem + wave * NPAIR * CCH;

    int  inidx[NPAIR], outidx[NPAIR];
    bool val[NPAIR];
#pragma unroll
    for (int i = 0; i < NPAIR; ++i) {
        const int p  = p0 + i;
        const int pc = (p < P) ? p : (P - 1);            // clamp tail for loads only
        val[i]    = (p < n);                             // n <= P-1, so val implies p < P
        inidx[i]  = pk[pc];                              // always a valid features row
        outidx[i] = pk[P + pc];
    }

#if USE_ASYNC
    // Stage 4 feature rows into LDS via the gfx1250 async copy engine.
    // Each lane moves 8 bytes -> one b64 per pair covers the full 256B row.
#pragma unroll
    for (int i = 0; i < NPAIR; ++i) {
        float* g = const_cast<float*>(feat) + (size_t)inidx[i] * CCH + lane * 2;
        float* l = wsm + i * CCH + lane * 2;
        __builtin_amdgcn_global_load_async_to_lds_b64(
            (gv2i_p)(void*)g,
            (lv2i_p)(void*)l,
            /*offset=*/0, /*cpol=*/0);
    }
    // Drain in order: async loads complete in issue order, so staggered waits
    // overlap pair i's atomics with pairs i+1..3 still in flight.
    __builtin_amdgcn_s_wait_asynccnt(3);
    asm volatile("" ::: "memory");
    spavg_process_pair(wsm + 0 * CCH + lane * 2, val[0], lane, outidx[0], out, counts);
    __builtin_amdgcn_s_wait_asynccnt(2);
    asm volatile("" ::: "memory");
    spavg_process_pair(wsm + 1 * CCH + lane * 2, val[1], lane, outidx[1], out, counts);
    __builtin_amdgcn_s_wait_asynccnt(1);
    asm volatile("" ::: "memory");
    spavg_process_pair(wsm + 2 * CCH + lane * 2, val[2], lane, outidx[2], out, counts);
    __builtin_amdgcn_s_wait_asynccnt(0);
    asm volatile("" ::: "memory");
    spavg_process_pair(wsm + 3 * CCH + lane * 2, val[3], lane, outidx[3], out, counts);
#else
    // Fallback: direct coalesced gather (global_load_b64 per lane) + atomics.
#pragma unroll
    for (int i = 0; i < NPAIR; ++i) {
        if (val[i]) {
            const float2 v = *(const float2*)(feat + (size_t)inidx[i] * CCH + lane * 2);
            float* dst = out + (size_t)outidx[i] * CCH + lane * 2;
            atomicAdd(dst,     v.x);
            atomicAdd(dst + 1, v.y);
            if (lane == 0) atomicAdd(counts + outidx[i], 1.0f);
        }
    }
    (void)wsm;
#endif
}

// ---------------- divide by count ----------------
__global__ void spavg_finalize_kernel(float4* __restrict__ out4,
                                      const float* __restrict__ cnt, int n4) {
    const int i = blockIdx.x * blockDim.x + threadIdx.x;
    if (i >= n4) return;
    const int row = i >> 4;                              // 16 float4 per 64-ch row
    const float s = 1.0f / fmaxf(cnt[row], 1.0f);
    float4 v = out4[i];
    v.x *= s; v.y *= s; v.z *= s; v.w *= s;
    out4[i] = v;
}

extern "C" void kernel_launch(void* const* d_in, const int* in_sizes, int n_in,
                              void* d_out, int out_size, void* d_ws, size_t ws_size,
                              hipStream_t stream) {
    const float* feat  = (const float*)d_in[0];   // (N_IN, 64) f32
    const int*   pairs = (const int*)  d_in[1];   // (K, 2, P) i32
    const int*   nums  = (const int*)  d_in[2];   // (K,) i32
    // d_in[3] = num_out scalar on device; recover on host from out_size.

    const int K       = in_sizes[2];
    const int P       = in_sizes[1] / (2 * K);
    const int num_out = out_size / CCH;

    float* out    = (float*)d_out;
    float* counts = (float*)d_ws;                 // num_out floats (~800 KB) of scratch

    const int n4 = out_size / 4;

    spavg_zero_kernel<<<4096, 256, 0, stream>>>((float4*)out, n4, counts, num_out);

    dim3 grid((P + WAVES * NPAIR - 1) / (WAVES * NPAIR), K);
    spavg_scatter_kernel<<<grid, WAVES * 32, 0, stream>>>(feat, pairs, nums, out, counts, P);

    const int fb = (n4 + 255) / 256;
    spavg_finalize_kernel<<<fb, 256, 0, stream>>>((float4*)out, counts, n4);
}